// PPN_56985626083854
// MI455X (gfx1250) — compile-verified
//
#include <hip/hip_runtime.h>

// ---------------- CDNA5 WMMA types ----------------
typedef float v2f __attribute__((ext_vector_type(2)));
typedef float v8f __attribute__((ext_vector_type(8)));

// ---------------- small utility kernels ----------------
__global__ void k_fill_i32(int* p, int n, int v) {
    int i = blockIdx.x * blockDim.x + threadIdx.x;
    if (i < n) p[i] = v;
}

__global__ void k_scatter(int* g, const int* __restrict__ coords, int N, int S) {
    int n = blockIdx.x * blockDim.x + threadIdx.x;
    if (n >= N) return;
    int x = coords[n * 3 + 0], y = coords[n * 3 + 1], z = coords[n * 3 + 2];
    g[(x * S + y) * S + z] = n;
}

__global__ void k_copy_f32(float* __restrict__ dst, const float* __restrict__ src, long n) {
    long i = (long)blockIdx.x * blockDim.x + threadIdx.x;
    if (i < n) dst[i] = src[i];
}

__global__ void k_add_f32(float* __restrict__ dst, const float* __restrict__ src, long n) {
    long i = (long)blockIdx.x * blockDim.x + threadIdx.x;
    if (i < n) dst[i] += src[i];
}

// ---------------- BN (batch stats over active sites) ----------------
// one block per channel; deterministic tree reduction (no float atomics)
__global__ void k_bn_stats(const float* __restrict__ x, const float* __restrict__ g,
                           const float* __restrict__ bt, float* __restrict__ sb,
                           int N, int C) {
    int c = blockIdx.x;
    __shared__ float ss[256], sq[256];
    float s = 0.f, q = 0.f;
    for (int i = threadIdx.x; i < N; i += 256) {
        float v = x[(long)i * C + c];
        s += v; q += v * v;
    }
    ss[threadIdx.x] = s; sq[threadIdx.x] = q;
    __syncthreads();
    for (int st = 128; st > 0; st >>= 1) {
        if ((int)threadIdx.x < st) {
            ss[threadIdx.x] += ss[threadIdx.x + st];
            sq[threadIdx.x] += sq[threadIdx.x + st];
        }
        __syncthreads();
    }
    if (threadIdx.x == 0) {
        float mu  = ss[0] / (float)N;
        float var = sq[0] / (float)N - mu * mu;
        float sc  = g[c] * rsqrtf(var + 1e-4f);
        sb[c]      = sc;              // scale
        sb[80 + c] = bt[c] - mu * sc; // bias
    }
}

__global__ void k_bnrelu(float* __restrict__ out, const float* __restrict__ x,
                         const float* __restrict__ sb, long total, int C) {
    long i = (long)blockIdx.x * blockDim.x + threadIdx.x;
    if (i >= total) return;
    int c = (int)(i % C);
    float v = x[i] * sb[c] + sb[80 + c];
    out[i] = v > 0.f ? v : 0.f;
}

// ---------------- gather-conv via f32 WMMA, LDS-staged B ----------------
// mode 0: submanifold 3x3x3 (27 taps, same level)
// mode 1: kernel-2 stride-2 down conv (8 taps, grid/S are the FINE level)
// Block = 4 waves; each wave owns a 16-row tile, all share one 16-out-channel
// chunk whose B tile is staged in LDS per tap (4x reuse, ds_load path).
__global__ __launch_bounds__(128)
void k_conv_wmma(const float* __restrict__ feat, const int* __restrict__ coords,
                 const int* __restrict__ grid, const float* __restrict__ W,
                 float* __restrict__ out, int N, int a, int b, int S, int mode) {
    extern __shared__ float ldsB[]; // a * 16 floats
    const int tid  = threadIdx.x;
    const int lane = tid & 31;
    const int wid  = tid >> 5;             // wave id 0..3
    const int m    = lane & 15;
    const int half = lane >> 4;            // 0 or 1
    const int row  = blockIdx.x * 64 + wid * 16 + m;
    const int rc   = row < N ? row : N - 1;
    const bool rvalid = row < N;
    const int cx = coords[rc * 3 + 0];
    const int cy = coords[rc * 3 + 1];
    const int cz = coords[rc * 3 + 2];
    const int bbase = blockIdx.y * 16;
    const int bcol  = bbase + m;           // output channel for B/D tiles
    const int K = (mode == 0) ? 27 : 8;
    const int ksel = half * 2;             // A layout: lanes<16 -> K=0,1 ; lanes>=16 -> K=2,3
    const int a16 = a * 16;

    v8f acc = {};
    for (int k = 0; k < K; ++k) {
        const float* Wk = W + (long)k * a * b;

        // ---- stage B tile  W[k][0:a, bbase:bbase+16] -> LDS (coalesced) ----
        for (int i = tid; i < a16; i += 128) {
            int cc = i >> 4, col = i & 15;
            ldsB[i] = Wk[(long)cc * b + bbase + col];
        }
        // prefetch next tap's weight slice while this tap computes
        if (k + 1 < K)
            __builtin_prefetch((const char*)(W + (long)(k + 1) * a * b) + tid * 64, 0, 1);
        __syncthreads();

        // ---- neighbor lookup for this tap ----
        int nx, ny, nz; bool v;
        if (mode == 0) {
            nx = cx + (k / 9) - 1;
            ny = cy + (k / 3) % 3 - 1;
            nz = cz + (k % 3) - 1;
            v = (nx >= 0) & (nx < S) & (ny >= 0) & (ny < S) & (nz >= 0) & (nz < S);
        } else {
            nx = cx * 2 + ((k >> 2) & 1);
            ny = cy * 2 + ((k >> 1) & 1);
            nz = cz * 2 + (k & 1);
            v = (nx < S) & (ny < S) & (nz < S);
        }
        int gx = v ? nx : 0, gy = v ? ny : 0, gz = v ? nz : 0;
        int srow = grid[(gx * S + gy) * S + gz];
        v = v && (srow >= 0) && rvalid;
        srow = srow >= 0 ? srow : 0;
        const float* frow = feat + (long)srow * a;

        // ---- K-chunks of 4: A gathered from global, B from LDS ----
#pragma unroll 4
        for (int c0 = 0; c0 < a; c0 += 4) {
            int cA = c0 + ksel;
            float a0 = (v && cA     < a) ? frow[cA]     : 0.f;
            float a1 = (v && cA + 1 < a) ? frow[cA + 1] : 0.f;
            float b0 = (cA     < a) ? ldsB[cA * 16 + m]       : 0.f;
            float b1 = (cA + 1 < a) ? ldsB[(cA + 1) * 16 + m] : 0.f;
            v2f av = {a0, a1};
            v2f bv = {b0, b1};
            // D = A(16x4 f32) * B(4x16 f32) + C  -- v_wmma_f32_16x16x4_f32
            acc = __builtin_amdgcn_wmma_f32_16x16x4_f32(
                false, av, false, bv, (short)0, acc, false, false);
        }
        __syncthreads(); // protect LDS before next tap's staging
    }
    // D layout: VGPR i holds M = i (lanes 0-15) / M = i+8 (lanes 16-31), N = lane&15
#pragma unroll
    for (int i = 0; i < 8; ++i) {
        int r = blockIdx.x * 64 + wid * 16 + i + half * 8;
        if (r < N) out[(long)r * b + bcol] = acc[i];
    }
}

// ---------------- 1x1 conv to 2 scores ----------------
__global__ void k_matmul2(const float* __restrict__ x, const float* __restrict__ sw,
                          float* __restrict__ out, int N, int C) {
    int n = blockIdx.x * blockDim.x + threadIdx.x;
    if (n >= N) return;
    float s0 = 0.f, s1 = 0.f;
    for (int c = 0; c < C; ++c) {
        float v = x[(long)n * C + c];
        s0 += v * sw[c * 2 + 0];
        s1 += v * sw[c * 2 + 1];
    }
    out[n * 2 + 0] = s0;
    out[n * 2 + 1] = s1;
}

// ---------------- unpool attention * feature ----------------
// selection: softmax([s0,s1])[1] > 0.5  <=>  s1 > s0
__global__ void k_attmul(float* __restrict__ out, const float* __restrict__ fm,
                         const int* __restrict__ coords_f, const int* __restrict__ grid_c,
                         const float* __restrict__ sc, long total, int C, int Sc, int shift) {
    long i = (long)blockIdx.x * blockDim.x + threadIdx.x;
    if (i >= total) return;
    int n = (int)(i / C);
    int cx = coords_f[n * 3 + 0] >> shift;
    int cy = coords_f[n * 3 + 1] >> shift;
    int cz = coords_f[n * 3 + 2] >> shift;
    int p = grid_c[(cx * Sc + cy) * Sc + cz];
    float att = (p >= 0 && sc[p * 2 + 1] > sc[p * 2 + 0]) ? 1.f : 0.f;
    out[i] = fm[i] * att;
}

// ---------------- output assembly ----------------
__global__ void k_out0(float* __restrict__ out, const float* __restrict__ z,
                       const float* __restrict__ pw, const float* __restrict__ sw, int N) {
    int n = blockIdx.x * blockDim.x + threadIdx.x;
    if (n >= N) return;
    float r0 = 0.f, r1 = 0.f, r2 = 0.f, r3 = 0.f, r4 = 0.f;
    for (int c = 0; c < 16; ++c) {
        float v = z[(long)n * 16 + c];
        r0 += v * pw[c * 3 + 0];
        r1 += v * pw[c * 3 + 1];
        r2 += v * pw[c * 3 + 2];
        r3 += v * sw[c * 2 + 0];
        r4 += v * sw[c * 2 + 1];
    }
    out[n * 5 + 0] = r0; out[n * 5 + 1] = r1; out[n * 5 + 2] = r2;
    out[n * 5 + 3] = r3; out[n * 5 + 4] = r4;
}

__global__ void k_outc(float* __restrict__ out, const int* __restrict__ coords,
                       const float* __restrict__ sc, int N) {
    int n = blockIdx.x * blockDim.x + threadIdx.x;
    if (n >= N) return;
    out[n * 5 + 0] = (float)coords[n * 3 + 0];
    out[n * 5 + 1] = (float)coords[n * 3 + 1];
    out[n * 5 + 2] = (float)coords[n * 3 + 2];
    out[n * 5 + 3] = sc[n * 2 + 0];
    out[n * 5 + 4] = sc[n * 2 + 1];
}

// ---------------- host side ----------------
extern "C" void kernel_launch(void* const* d_in, const int* in_sizes, int n_in,
                              void* d_out, int out_size, void* d_ws, size_t ws_size,
                              hipStream_t stream) {
    static const int CL[5]  = {16, 32, 48, 64, 80};
    static const int SLl[5] = {192, 96, 48, 24, 12};

    const int* coords[5];
    int Nl[5];
    for (int l = 0; l < 5; ++l) {
        coords[l] = (const int*)d_in[l];
        Nl[l] = in_sizes[l] / 3;
    }
    const float* features = (const float*)d_in[5];
    const float* w_in     = (const float*)d_in[6];
    const float *blk_w[4], *blk_g[4], *blk_b[4], *dn_g[4], *dn_b[4], *dn_w[4];
    for (int i = 0; i < 4; ++i) {
        int base = 7 + i * 6;
        blk_w[i] = (const float*)d_in[base + 0];
        blk_g[i] = (const float*)d_in[base + 1];
        blk_b[i] = (const float*)d_in[base + 2];
        dn_g[i]  = (const float*)d_in[base + 3];
        dn_b[i]  = (const float*)d_in[base + 4];
        dn_w[i]  = (const float*)d_in[base + 5];
    }
    const float* ppn1_w  = (const float*)d_in[31];
    const float* ppn1_sw = (const float*)d_in[32];
    const float* ppn2_w  = (const float*)d_in[33];
    const float* ppn2_sw = (const float*)d_in[34];
    const float* ppn3_w  = (const float*)d_in[35];
    const float* ppn3_pw = (const float*)d_in[36];
    const float* ppn3_sw = (const float*)d_in[37];
    float* out = (float*)d_out;

    // ---- carve workspace ----
    char* ws = (char*)d_ws;
    size_t off = 0;
    auto alloc = [&](size_t bytes) -> void* {
        void* p = ws + off;
        off = (off + bytes + 255) & ~(size_t)255;
        return p;
    };
    int* grids[5]; long gsz[5];
    for (int l = 0; l < 5; ++l) {
        gsz[l] = (long)SLl[l] * SLl[l] * SLl[l];
        grids[l] = (int*)alloc((size_t)gsz[l] * 4);
    }
    float* fm[5];
    for (int l = 0; l < 5; ++l) fm[l] = (float*)alloc((size_t)Nl[l] * CL[l] * 4);
    size_t maxNC = 0;
    for (int l = 0; l < 5; ++l) {
        size_t nc = (size_t)Nl[l] * CL[l];
        if (nc > maxNC) maxNC = nc;
    }
    float* xw = (float*)alloc(maxNC * 4);
    float* tA = (float*)alloc(maxNC * 4);
    float* tB = (float*)alloc(maxNC * 4);
    float* sb = (float*)alloc(160 * 4);
    float* s1 = (float*)alloc((size_t)Nl[4] * 2 * 4);
    float* s2 = (float*)alloc((size_t)Nl[2] * 2 * 4);
    if (off > ws_size) return; // scratch insufficient (should not happen)

    const int TB = 256;
    auto blks = [&](long n) { return (int)((n + TB - 1) / TB); };
    auto conv = [&](const float* f, const int* co, const int* gr, const float* W,
                    float* o, int N, int a, int b, int S, int mode) {
        dim3 g((N + 63) / 64, b / 16);
        size_t shm = (size_t)a * 16 * sizeof(float);
        k_conv_wmma<<<g, 128, shm, stream>>>(f, co, gr, W, o, N, a, b, S, mode);
    };

    // ---- grids ----
    for (int l = 0; l < 5; ++l) {
        k_fill_i32<<<blks(gsz[l]), TB, 0, stream>>>(grids[l], (int)gsz[l], -1);
        k_scatter<<<blks(Nl[l]), TB, 0, stream>>>(grids[l], coords[l], Nl[l], SLl[l]);
    }

    // ---- input conv: [N0,1] x w_in[27,1,16] -> fm0 ----
    conv(features, coords[0], grids[0], w_in, fm[0], Nl[0], 1, 16, SLl[0], 0);
    k_copy_f32<<<blks((long)Nl[0] * 16), TB, 0, stream>>>(xw, fm[0], (long)Nl[0] * 16);

    // ---- encoder: 4 levels, 2 residual blocks + BN+ReLU + downconv ----
    for (int i = 0; i < 4; ++i) {
        int N = Nl[i], C = CL[i], S = SLl[i];
        long nc = (long)N * C;
        const int* co = coords[i];
        const int* gr = grids[i];
        for (int r = 0; r < 2; ++r) {
            size_t woff = (size_t)(2 * r) * 27 * C * C;
            k_bn_stats<<<C, TB, 0, stream>>>(xw, blk_g[i] + (2 * r) * C, blk_b[i] + (2 * r) * C, sb, N, C);
            k_bnrelu<<<blks(nc), TB, 0, stream>>>(tA, xw, sb, nc, C);
            conv(tA, co, gr, blk_w[i] + woff, tB, N, C, C, S, 0);
            k_bn_stats<<<C, TB, 0, stream>>>(tB, blk_g[i] + (2 * r + 1) * C, blk_b[i] + (2 * r + 1) * C, sb, N, C);
            k_bnrelu<<<blks(nc), TB, 0, stream>>>(tA, tB, sb, nc, C);
            conv(tA, co, gr, blk_w[i] + woff + (size_t)27 * C * C, tB, N, C, C, S, 0);
            k_add_f32<<<blks(nc), TB, 0, stream>>>(xw, tB, nc);
        }
        // BN+ReLU then stride-2 down conv to level i+1
        k_bn_stats<<<C, TB, 0, stream>>>(xw, dn_g[i], dn_b[i], sb, N, C);
        k_bnrelu<<<blks(nc), TB, 0, stream>>>(tA, xw, sb, nc, C);
        int No = Nl[i + 1], Co = CL[i + 1];
        conv(tA, coords[i + 1], gr, dn_w[i], fm[i + 1], No, C, Co, S, 1);
        k_copy_f32<<<blks((long)No * Co), TB, 0, stream>>>(xw, fm[i + 1], (long)No * Co);
    }

    // ---- PPN head 1 at level 4 ----
    conv(xw, coords[4], grids[4], ppn1_w, tB, Nl[4], 80, 80, SLl[4], 0);
    k_matmul2<<<blks(Nl[4]), TB, 0, stream>>>(tB, ppn1_sw, s1, Nl[4], 80);

    // ---- attention 1 -> level 2, PPN head 2 ----
    {
        long nc2 = (long)Nl[2] * 48;
        k_attmul<<<blks(nc2), TB, 0, stream>>>(tA, fm[2], coords[2], grids[4], s1, nc2, 48, SLl[4], 2);
        conv(tA, coords[2], grids[2], ppn2_w, tB, Nl[2], 48, 48, SLl[2], 0);
        k_matmul2<<<blks(Nl[2]), TB, 0, stream>>>(tB, ppn2_sw, s2, Nl[2], 48);
    }

    // ---- attention 2 -> level 0, PPN head 3 ----
    {
        long nc0 = (long)Nl[0] * 16;
        k_attmul<<<blks(nc0), TB, 0, stream>>>(tA, fm[0], coords[0], grids[2], s2, nc0, 16, SLl[2], 2);
        conv(tA, coords[0], grids[0], ppn3_w, tB, Nl[0], 16, 16, SLl[0], 0);
    }

    // ---- outputs: out0 [N0,5] ++ out1 [N4,5] ++ out2 [N2,5] ----
    k_out0<<<blks(Nl[0]), TB, 0, stream>>>(out, tB, ppn3_pw, ppn3_sw, Nl[0]);
    k_outc<<<blks(Nl[4]), TB, 0, stream>>>(out + (size_t)5 * Nl[0], coords[4], s1, Nl[4]);
    k_outc<<<blks(Nl[2]), TB, 0, stream>>>(out + (size_t)5 * (Nl[0] + Nl[4]), coords[2], s2, Nl[2]);
}